// GCNModel_44220983280095
// MI455X (gfx1250) — compile-verified
//
#include <hip/hip_runtime.h>
#include <hip/hip_bf16.h>

typedef __attribute__((ext_vector_type(2))) float v2f;
typedef __attribute__((ext_vector_type(8))) float v8f;

#define IN_DIM 512
#define HID 256
#define KOUT 16
#define NEG_SLOPE 0.2f

// ---------------- degree / norm ----------------

__global__ void k_fill1(float* __restrict__ p, int n) {
  int i = blockIdx.x * blockDim.x + threadIdx.x;
  if (i < n) p[i] = 1.0f;  // self-loop contributes 1 to every node's degree
}

__global__ void k_deg_accum(const long long* __restrict__ dst,
                            float* __restrict__ deg, int E) {
  int e = blockIdx.x * blockDim.x + threadIdx.x;
  if (e < E) atomicAdd(&deg[(int)dst[e]], 1.0f);
}

__global__ void k_rsqrt_inplace(float* __restrict__ p, int n) {
  int i = blockIdx.x * blockDim.x + threadIdx.x;
  if (i < n) p[i] = rsqrtf(p[i]);  // deg >= 1 always (self-loops)
}

// ---- fp32 WMMA GEMM, N-blocked: one wave computes a 16 x (16*NT) C strip ----
// A-frag 16x4 (v2f/lane): lanes 0-15 hold K=k,k+1 of row m0+lane;
//                         lanes 16-31 hold K=k+2,k+3.  Reused across NT tiles.
// B-frag 4x16 (v2f/lane): lanes 0-15 rows k,k+1 at col n0+lane;
//                         lanes 16-31 rows k+2,k+3.
// C/D 8 VGPRs: VGPR r = row m0+r (lanes 0-15) / m0+r+8 (lanes 16-31).
// Epilogue fuses the GCN self-loop term: Craw = A@B, Cscl = dinv[row]^2 * (A@B).

template <int NT>
__global__ void k_gemm_wmma_f32(const float* __restrict__ A,
                                const float* __restrict__ B,
                                float* __restrict__ Craw,
                                float* __restrict__ Cscl,
                                const float* __restrict__ dinv,
                                int M, int Kd, int Nc) {
  int wave = (int)((blockIdx.x * blockDim.x + threadIdx.x) >> 5);
  int lane = threadIdx.x & 31;
  int nsuper = Nc / (16 * NT);
  int total = (M >> 4) * nsuper;
  if (wave >= total) return;            // wave-uniform: EXEC stays all-ones

  int tm = wave / nsuper;
  int tn = wave - tm * nsuper;
  int m0 = tm << 4;
  int n0 = tn * (16 * NT);
  int half = lane >> 4;                 // 0: K pair (k,k+1); 1: (k+2,k+3)
  int lr = lane & 15;

  const float* arow = A + (size_t)(m0 + lr) * Kd + (half << 1);
  const float* brow = B + (size_t)(half << 1) * Nc + n0 + lr;

  v8f acc[NT];
#pragma unroll
  for (int t = 0; t < NT; ++t) acc[t] = (v8f){};

#pragma unroll 2
  for (int k = 0; k < Kd; k += 4) {
    v2f af = *(const v2f*)(arow + k);   // 64-bit load, reused across NT wmmas
#pragma unroll
    for (int t = 0; t < NT; ++t) {
      v2f bf;
      bf.x = brow[(size_t)k * Nc + t * 16];
      bf.y = brow[(size_t)(k + 1) * Nc + t * 16];
      acc[t] = __builtin_amdgcn_wmma_f32_16x16x4_f32(
          /*neg_a=*/false, af, /*neg_b=*/false, bf,
          /*c_mod=*/(short)0, acc[t], /*reuse_a=*/false, /*reuse_b=*/false);
    }
  }

#pragma unroll
  for (int r = 0; r < 8; ++r) {
    int row = m0 + r + (half << 3);
    float di = dinv[row];               // wave-half-uniform broadcast load
    float sc = di * di;
#pragma unroll
    for (int t = 0; t < NT; ++t) {
      size_t idx = (size_t)row * Nc + n0 + t * 16 + lr;
      float v = acc[t][r];
      Craw[idx] = v;                    // gather source for message passing
      Cscl[idx] = sc * v;               // fused self-loop init of the scatter dest
    }
  }
}

// ---------------- edge scatter (F==256): one wave per edge, float4 ----------------

__global__ void k_edge_agg256(const long long* __restrict__ src,
                              const long long* __restrict__ dst,
                              const float* __restrict__ dinv,
                              const float* __restrict__ H,
                              float* __restrict__ Out, int E) {
  int wave = (int)((blockIdx.x * blockDim.x + threadIdx.x) >> 5);
  int lane = threadIdx.x & 31;
  if (wave >= E) return;
  int s = (int)src[wave], d = (int)dst[wave];
  float nrm = dinv[s] * dinv[d];
  const float4* hs = (const float4*)(H + (size_t)s * 256);
  float* od = Out + (size_t)d * 256;
#pragma unroll
  for (int it = 0; it < 2; ++it) {
    int f4 = lane + it * 32;            // float4 index 0..63 -> b128 coalesced
    float4 v = hs[f4];
    atomicAdd(od + f4 * 4 + 0, nrm * v.x);
    atomicAdd(od + f4 * 4 + 1, nrm * v.y);
    atomicAdd(od + f4 * 4 + 2, nrm * v.z);
    atomicAdd(od + f4 * 4 + 3, nrm * v.w);
  }
}

// ---------------- edge scatter (F==16): one thread per (edge, 4 feats) ----------------

__global__ void k_edge_agg16(const long long* __restrict__ src,
                             const long long* __restrict__ dst,
                             const float* __restrict__ dinv,
                             const float* __restrict__ H,
                             float* __restrict__ Out, int E) {
  int i = blockIdx.x * blockDim.x + threadIdx.x;
  int e = i >> 2;
  if (e >= E) return;
  int q = i & 3;                        // which float4 of the 16-wide row
  int s = (int)src[e], d = (int)dst[e];
  float nrm = dinv[s] * dinv[d];
  float4 v = ((const float4*)(H + (size_t)s * 16))[q];
  float* od = Out + (size_t)d * 16 + q * 4;
  atomicAdd(od + 0, nrm * v.x);
  atomicAdd(od + 1, nrm * v.y);
  atomicAdd(od + 2, nrm * v.z);
  atomicAdd(od + 3, nrm * v.w);
}

// ---------------- bias + LeakyReLU(0.2) on [N, 256], float4 ----------------

__global__ void k_bias_lrelu256(float* __restrict__ A, const float* __restrict__ b,
                                int total4) {
  int i = blockIdx.x * blockDim.x + threadIdx.x;
  if (i >= total4) return;
  float4 v = ((const float4*)A)[i];
  float4 bb = ((const float4*)b)[i & 63];   // 256/4 = 64 float4 of bias
  v.x += bb.x; v.y += bb.y; v.z += bb.z; v.w += bb.w;
  v.x = v.x >= 0.0f ? v.x : NEG_SLOPE * v.x;
  v.y = v.y >= 0.0f ? v.y : NEG_SLOPE * v.y;
  v.z = v.z >= 0.0f ? v.z : NEG_SLOPE * v.z;
  v.w = v.w >= 0.0f ? v.w : NEG_SLOPE * v.w;
  ((float4*)A)[i] = v;
}

// ---------------- bias + softmax over K=16, one thread per row ----------------

__global__ void k_bias_softmax16(float* __restrict__ O,
                                 const float* __restrict__ b2, int n) {
  int i = blockIdx.x * blockDim.x + threadIdx.x;
  if (i >= n) return;
  float* row = O + (size_t)i * 16;
  float v[16];
  float m = -__builtin_inff();
#pragma unroll
  for (int f = 0; f < 16; ++f) { v[f] = row[f] + b2[f]; m = fmaxf(m, v[f]); }
  float s = 0.0f;
#pragma unroll
  for (int f = 0; f < 16; ++f) { v[f] = __expf(v[f] - m); s += v[f]; }
  float inv = 1.0f / s;
#pragma unroll
  for (int f = 0; f < 16; ++f) row[f] = v[f] * inv;
}

// ---------------- launch ----------------

extern "C" void kernel_launch(void* const* d_in, const int* in_sizes, int n_in,
                              void* d_out, int out_size, void* d_ws, size_t ws_size,
                              hipStream_t stream) {
  const float* x      = (const float*)d_in[0];
  const long long* ei = (const long long*)d_in[1];   // int64 [2, E]
  const float* W1     = (const float*)d_in[2];
  const float* b1     = (const float*)d_in[3];
  const float* W2     = (const float*)d_in[4];
  const float* b2     = (const float*)d_in[5];
  float* out          = (float*)d_out;

  const int N = in_sizes[0] / IN_DIM;   // 50000 (multiple of 16)
  const int E = in_sizes[1] / 2;        // 800000
  const long long* src = ei;
  const long long* dst = ei + E;

  // workspace layout (floats): dinv | H1[N,HID] | A1[N,HID] | H2[N,KOUT]
  float* ws   = (float*)d_ws;
  size_t off  = 0;
  float* dinv = ws + off; off += (size_t)((N + 15) & ~15);
  float* H1   = ws + off; off += (size_t)N * HID;
  float* A1   = ws + off; off += (size_t)N * HID;
  float* H2   = ws + off;

  const int T = 256;

  // 1) deg = 1 + sum over dst;  dinv = rsqrt(deg)
  k_fill1<<<(N + T - 1) / T, T, 0, stream>>>(dinv, N);
  k_deg_accum<<<(E + T - 1) / T, T, 0, stream>>>(dst, dinv, E);
  k_rsqrt_inplace<<<(N + T - 1) / T, T, 0, stream>>>(dinv, N);

  // 2) H1 = x @ W1 ; A1 = dinv^2 * H1   (fp32 WMMA, 16x64 strip per wave)
  {
    int waves = (N >> 4) * (HID / 64);  // 3125 * 4
    k_gemm_wmma_f32<4><<<(waves + 7) / 8, T, 0, stream>>>(
        x, W1, H1, A1, dinv, N, IN_DIM, HID);
  }

  // 3) A1 += sum_e norm_e * H1[src]
  k_edge_agg256<<<((size_t)E * 32 + T - 1) / T, T, 0, stream>>>(
      src, dst, dinv, H1, A1, E);

  // 4) bias + LeakyReLU
  k_bias_lrelu256<<<(N * HID / 4 + T - 1) / T, T, 0, stream>>>(A1, b1, N * HID / 4);

  // 5) H2 = A1 @ W2 ; out = dinv^2 * H2   (fp32 WMMA, single N tile)
  {
    int waves = (N >> 4);               // 3125
    k_gemm_wmma_f32<1><<<(waves + 7) / 8, T, 0, stream>>>(
        A1, W2, H2, out, dinv, N, HID, KOUT);
  }

  // 6) out += sum_e norm_e * H2[src]
  k_edge_agg16<<<((size_t)E * 4 + T - 1) / T, T, 0, stream>>>(
      src, dst, dinv, H2, out, E);

  // 7) bias + softmax over the 16 classes
  k_bias_softmax16<<<(N + T - 1) / T, T, 0, stream>>>(out, b2, N);
}